// Gating_91190745629222
// MI455X (gfx1250) — compile-verified
//
#include <hip/hip_runtime.h>
#include <hip/hip_bf16.h>

#define S_TOK 16384
#define D_DIM 4096
#define E_EXP 64
#define CAP   320      // int(1.25 * 16384 / 64)
#define NB    64       // blocks of 256 tokens for the dispatch passes
#define KC    32       // K-chunk staged in LDS

typedef __attribute__((ext_vector_type(2))) float v2f;
typedef __attribute__((ext_vector_type(8))) float v8f;
typedef __attribute__((ext_vector_type(4))) unsigned int u32x4;
typedef __attribute__((ext_vector_type(8))) int i32x8;
typedef __attribute__((ext_vector_type(4))) int i32x4;

// ---------------------------------------------------------------------------
// Tensor Data Mover: 2D tile (tile_w x tile_h elements, fp32) global -> LDS.
// Descriptor layout per CDNA5 ISA ch.8 (group0 128b, group1 256b).
// This toolchain exposes the 6-arg builtin: (g0, g1, g2, g3, g4, cpol).
// ---------------------------------------------------------------------------
__device__ __forceinline__ void tdm_load_2d_f32(
    const float* gptr, unsigned lds_addr,
    unsigned tile_w, unsigned tile_h,
    unsigned tensor_w, unsigned tensor_h, unsigned stride_elems)
{
    unsigned long long ga = (unsigned long long)(uintptr_t)gptr;
    u32x4 g0;
    g0[0] = 1u;                                            // count=1, user D#
    g0[1] = lds_addr;                                      // lds_addr[31:0]
    g0[2] = (unsigned)(ga & 0xffffffffu);                  // global_addr[31:0]
    g0[3] = (unsigned)((ga >> 32) & 0x1ffffffu)            // global_addr[56:32]
          | (2u << 30);                                    // type=2 ("image")
    i32x8 g1;
    g1[0] = (int)(2u << 16);                               // data_size=2 (4B)
    g1[1] = (int)((tensor_w & 0xffffu) << 16);             // tensor_dim0[15:0]
    g1[2] = (int)((tensor_w >> 16) | ((tensor_h & 0xffffu) << 16));
    g1[3] = (int)((tensor_h >> 16) | (tile_w << 16));      // tile_dim0
    g1[4] = (int)(tile_h & 0xffffu);                       // tile_dim1; tile_dim2=0
    g1[5] = (int)stride_elems;                             // tensor_dim0_stride[31:0]
    g1[6] = 0;                                             // stride hi / dim1_stride lo
    g1[7] = 0;
    i32x4 z4 = {0, 0, 0, 0};
    i32x8 z8 = {0, 0, 0, 0, 0, 0, 0, 0};
    __builtin_amdgcn_tensor_load_to_lds(g0, g1, z4, z4, z8, 0);
}

// ---------------------------------------------------------------------------
// Kernel 1: logits = x @ W_g.T via V_WMMA_F32_16X16X4_F32 with double-buffered
// TDM tile loads, then softmax/top-2. 256 threads = 8 waves; WG = 128 tokens.
// ---------------------------------------------------------------------------
__global__ __launch_bounds__(256) void k_gemm_top2(
    const float* __restrict__ x, const float* __restrict__ Wg,
    int* __restrict__ e1a, int* __restrict__ e2a,
    float* __restrict__ g1a, float* __restrict__ g2a, int* __restrict__ w2a)
{
    // 48 KB: [xs0 16K][wsh0 8K][xs1 16K][wsh1 8K]; lg (32K) aliases the front.
    __shared__ __align__(16) char smem[49152];
    float* const xs0  = (float*)(smem);
    float* const wsh0 = (float*)(smem + 16384);
    float* const xs1  = (float*)(smem + 24576);
    float* const wsh1 = (float*)(smem + 40960);
    float* const lg   = (float*)(smem);        // staged logits (after K loop)

    const int tid  = threadIdx.x;
    const int lane = tid & 31;
    const int wv   = tid >> 5;
    const int lr   = lane & 15;                // row/col-in-tile lane
    const int half = lane >> 4;                // 0: K pair 0..1, 1: K pair 2..3
    const int tok0 = blockIdx.x * 128;

    // Wave-relative LDS byte offsets (flat LDS address truncates to low 32b).
    const unsigned xs_off[2]  = { (unsigned)(uintptr_t)xs0,  (unsigned)(uintptr_t)xs1  };
    const unsigned wsh_off[2] = { (unsigned)(uintptr_t)wsh0, (unsigned)(uintptr_t)wsh1 };

    v8f acc0 = {}, acc1 = {}, acc2 = {}, acc3 = {};

    const int NC = D_DIM / KC;                 // 128 chunks

    // Prologue: wave 0 DMAs chunk 0 into buffer 0 (x tile + W tile).
    if (wv == 0) {
        tdm_load_2d_f32(x + (size_t)tok0 * D_DIM, xs_off[0], KC, 128, D_DIM, S_TOK, D_DIM);
        tdm_load_2d_f32(Wg, wsh_off[0], KC, 64, D_DIM, E_EXP, D_DIM);
    }

    for (int c = 0; c < NC; ++c) {
        const int p = c & 1;
        if (wv == 0) {
            if (c + 1 < NC) {
                const int k0n = (c + 1) * KC;
                tdm_load_2d_f32(x + (size_t)tok0 * D_DIM + k0n, xs_off[p ^ 1],
                                KC, 128, D_DIM, S_TOK, D_DIM);
                tdm_load_2d_f32(Wg + k0n, wsh_off[p ^ 1], KC, 64, D_DIM, E_EXP, D_DIM);
                __builtin_amdgcn_s_wait_tensorcnt(2);   // chunk c landed (in-order)
            } else {
                __builtin_amdgcn_s_wait_tensorcnt(0);   // drain TDM
            }
        }
        __syncthreads();

        const float* const xsb = p ? xs1 : xs0;
        const float* const wsb = p ? wsh1 : wsh0;

        #pragma unroll
        for (int kk = 0; kk < KC; kk += 4) {
            // A fragment (16x4): lanes 0-15 rows, K in VGPR pair; lanes 16-31 K+2
            const float* ap = xsb + (wv * 16 + lr) * KC + kk + 2 * half;
            v2f a; a.x = ap[0]; a.y = ap[1];
            // B fragments (4x16), B[k][n] = Wg[n][k]; one per N-tile of 16 experts
            const float* bp0 = wsb + (0  + lr) * KC + kk + 2 * half;
            const float* bp1 = wsb + (16 + lr) * KC + kk + 2 * half;
            const float* bp2 = wsb + (32 + lr) * KC + kk + 2 * half;
            const float* bp3 = wsb + (48 + lr) * KC + kk + 2 * half;
            v2f b0; b0.x = bp0[0]; b0.y = bp0[1];
            v2f b1; b1.x = bp1[0]; b1.y = bp1[1];
            v2f b2; b2.x = bp2[0]; b2.y = bp2[1];
            v2f b3; b3.x = bp3[0]; b3.y = bp3[1];
            acc0 = __builtin_amdgcn_wmma_f32_16x16x4_f32(false, a, false, b0, (short)0, acc0, false, false);
            acc1 = __builtin_amdgcn_wmma_f32_16x16x4_f32(false, a, false, b1, (short)0, acc1, false, false);
            acc2 = __builtin_amdgcn_wmma_f32_16x16x4_f32(false, a, false, b2, (short)0, acc2, false, false);
            acc3 = __builtin_amdgcn_wmma_f32_16x16x4_f32(false, a, false, b3, (short)0, acc3, false, false);
        }
        __syncthreads();    // all waves done with buf p before it is re-filled
    }

    // Spill C fragments (lanes 0-15: M=j, lanes 16-31: M=j+8; N = lane&15) to LDS.
    #pragma unroll
    for (int j = 0; j < 8; ++j) {
        int r = j + 8 * half;
        lg[(wv * 16 + r) * 64 + 0  + lr] = acc0[j];
        lg[(wv * 16 + r) * 64 + 16 + lr] = acc1[j];
        lg[(wv * 16 + r) * 64 + 32 + lr] = acc2[j];
        lg[(wv * 16 + r) * 64 + 48 + lr] = acc3[j];
    }
    __syncthreads();

    // One thread per token: softmax over 64 experts, top-2, normalized gates.
    if (tid < 128) {
        const float* row = lg + tid * 64;
        float m = row[0];
        #pragma unroll 8
        for (int e = 1; e < 64; ++e) m = fmaxf(m, row[e]);
        float sum = 0.0f, b1 = -3.0e38f, b2 = -3.0e38f;
        int i1 = 0, i2 = 0;
        for (int e = 0; e < 64; ++e) {
            float v = row[e];
            sum += __expf(v - m);
            if (v > b1)      { b2 = b1; i2 = i1; b1 = v; i1 = e; }
            else if (v > b2) { b2 = v;  i2 = e; }
        }
        float g1 = __expf(b1 - m) / sum;
        float g2 = __expf(b2 - m) / sum;
        float den = g1 + g2;
        float g1n = g1 / den;
        float g2n = g2 / den;

        int t = tok0 + tid;
        // Deterministic per-token uniform (stands in for jax.random.uniform(key(1))).
        unsigned u = (unsigned)t;
        u ^= u >> 16; u *= 0x7feb352du;
        u ^= u >> 15; u *= 0x846ca68bu;
        u ^= u >> 16;
        float rnd = (float)(u >> 8) * (1.0f / 16777216.0f);
        int want2 = (rnd < 2.0f * g2n) ? 1 : 0;

        e1a[t] = i1; e2a[t] = i2;
        g1a[t] = g1n; g2a[t] = g2n;
        w2a[t] = want2;
    }
}

// ---------------------------------------------------------------------------
// Kernel 2: per-block expert histograms (primary; secondary gated by want2).
// ---------------------------------------------------------------------------
__global__ __launch_bounds__(256) void k_hist(
    const int* __restrict__ e1a, const int* __restrict__ e2a,
    const int* __restrict__ w2a, int* __restrict__ hist1, int* __restrict__ hist2)
{
    __shared__ int h1[E_EXP], h2[E_EXP];
    const int tid = threadIdx.x, b = blockIdx.x;
    if (tid < E_EXP) { h1[tid] = 0; h2[tid] = 0; }
    __syncthreads();
    const int t = b * 256 + tid;
    atomicAdd(&h1[e1a[t]], 1);
    if (w2a[t]) atomicAdd(&h2[e2a[t]], 1);
    __syncthreads();
    if (tid < E_EXP) {
        hist1[b * E_EXP + tid] = h1[tid];
        hist2[b * E_EXP + tid] = h2[tid];
    }
}

// ---------------------------------------------------------------------------
// Kernel 3: per-expert exclusive scan over blocks; residual capacity for slot 2.
// ---------------------------------------------------------------------------
__global__ __launch_bounds__(64) void k_scan(
    const int* __restrict__ hist1, const int* __restrict__ hist2,
    int* __restrict__ off1, int* __restrict__ off2, int* __restrict__ rem2)
{
    const int e = threadIdx.x;
    int run = 0;
    for (int b = 0; b < NB; ++b) { off1[b * E_EXP + e] = run; run += hist1[b * E_EXP + e]; }
    int c1 = run < CAP ? run : CAP;           // counts1[e] = min(total1, CAP)
    rem2[e] = CAP - c1;
    run = 0;
    for (int b = 0; b < NB; ++b) { off2[b * E_EXP + e] = run; run += hist2[b * E_EXP + e]; }
}

// ---------------------------------------------------------------------------
// Kernel 4: ordered ranks (ballot + cross-wave scan), capacity test, scatter.
// ---------------------------------------------------------------------------
__global__ __launch_bounds__(256) void k_finalize(
    const int* __restrict__ e1a, const int* __restrict__ e2a,
    const float* __restrict__ g1a, const float* __restrict__ g2a,
    const int* __restrict__ w2a,
    const int* __restrict__ off1, const int* __restrict__ off2,
    const int* __restrict__ rem2, float* __restrict__ out)
{
    __shared__ int c1[8][E_EXP], c2[8][E_EXP];
    const int tid = threadIdx.x, b = blockIdx.x;
    const int lane = tid & 31, wv = tid >> 5;
    const int t = b * 256 + tid;

    const int me1 = e1a[t];
    const int me2 = e2a[t];
    const int w2  = w2a[t];
    const unsigned below = (1u << lane) - 1u;

    int rank1 = 0, rank2 = 0;
    for (int e = 0; e < E_EXP; ++e) {
        unsigned m1 = (unsigned)__ballot(me1 == e);
        if (lane == 0) c1[wv][e] = __popc(m1);
        if (me1 == e)  rank1 = __popc(m1 & below);
        unsigned m2 = (unsigned)__ballot(w2 && me2 == e);
        if (lane == 0) c2[wv][e] = __popc(m2);
        if (w2 && me2 == e) rank2 = __popc(m2 & below);
    }
    __syncthreads();

    int base1 = 0, base2 = 0;
    for (int w = 0; w < wv; ++w) { base1 += c1[w][me1]; base2 += c2[w][me2]; }

    const int pos1 = off1[b * E_EXP + me1] + base1 + rank1 + 1; // 1-based, incl. self
    const int keep1 = (pos1 <= CAP);
    const int pos2 = off2[b * E_EXP + me2] + base2 + rank2 + 1;
    const int keep2 = w2 && (pos2 <= rem2[me2]);

    const float v1 = keep1 ? g1a[t] : 0.0f;
    const float v2 = keep2 ? g2a[t] : 0.0f;

    float* row = out + (size_t)t * E_EXP;
    #pragma unroll
    for (int e = 0; e < E_EXP; e += 4) {
        float4 v = make_float4(0.f, 0.f, 0.f, 0.f);
        if (me1 >= e && me1 < e + 4) ((float*)&v)[me1 - e] = v1;
        if (me2 >= e && me2 < e + 4) ((float*)&v)[me2 - e] = v2;
        *(float4*)(row + e) = v;
    }
}

// ---------------------------------------------------------------------------
extern "C" void kernel_launch(void* const* d_in, const int* in_sizes, int n_in,
                              void* d_out, int out_size, void* d_ws, size_t ws_size,
                              hipStream_t stream) {
    const float* x  = (const float*)d_in[0];   // [S, D] fp32
    const float* Wg = (const float*)d_in[1];   // [E, D] fp32
    float* out = (float*)d_out;                // [S, E] fp32

    // Workspace carve-up (~394 KB total)
    int*   e1a   = (int*)d_ws;                 // S
    int*   e2a   = e1a + S_TOK;                // S
    float* g1a   = (float*)(e2a + S_TOK);      // S
    float* g2a   = g1a + S_TOK;                // S
    int*   w2a   = (int*)(g2a + S_TOK);        // S
    int*   hist1 = w2a + S_TOK;                // NB*E
    int*   hist2 = hist1 + NB * E_EXP;         // NB*E
    int*   off1  = hist2 + NB * E_EXP;         // NB*E
    int*   off2  = off1 + NB * E_EXP;          // NB*E
    int*   rem2  = off2 + NB * E_EXP;          // E

    k_gemm_top2<<<S_TOK / 128, 256, 0, stream>>>(x, Wg, e1a, e2a, g1a, g2a, w2a);
    k_hist<<<NB, 256, 0, stream>>>(e1a, e2a, w2a, hist1, hist2);
    k_scan<<<1, 64, 0, stream>>>(hist1, hist2, off1, off2, rem2);
    k_finalize<<<NB, 256, 0, stream>>>(e1a, e2a, g1a, g2a, w2a, off1, off2, rem2, out);
}